// PrimalDualNetwork_2_18837726560653
// MI455X (gfx1250) — compile-verified
//
#include <hip/hip_runtime.h>
#include <math.h>

// Primal-dual TV-ROF (Chambolle-Pock), 2048x2048 fp32, 10 iterations.
// Working set (x, x_tilde, y[2], w[2], img) ~= 118 MB -> fully L2-resident
// on MI455X (192 MB L2). Bandwidth-bound stencil: b128 access, branchless
// boundary handling via address clamping (boundary values are exact zeros),
// wave32, 8 waves/block. WMMA does not apply (no matmul structure).
#define MM 2048
#define NN 2048
#define NC (NN / 4)               // float4 chunks per row (512, power of two)
#define NC_SHIFT 9
#define NC_MASK 511u
#define PLANE (MM * NN)
#define PLANE_BYTES ((size_t)PLANE * 4)  // 16 MiB per plane

// Compile-time literals -> inline VOP immediates.
constexpr float kSIGMA = (float)(1.0 / (7.0 * 0.01));  // 14.285714...
constexpr float kTAU   = 0.01f;
constexpr float kLT    = (float)(4.0 * 0.01);          // LAMBDA_ROF * TAU
constexpr float kDEN   = (float)(1.0 + 4.0 * 0.01);    // 1.04
constexpr float kTHETA = 0.5f;

static __device__ __forceinline__ float clamp1(float p) {
    // prox_linf_ball(p, 1) == clamp(p, -1, 1) in value
    return fminf(1.0f, fmaxf(-1.0f, p));
}

// ---------------------------------------------------------------------------
// init: x = xt = img ; y = forward_grad(img) ; w = w1 + w2*exp(-|y|)
// Branchless: right/bottom halo loads are clamped so the boundary difference
// is exactly 0 (self minus self).
// ---------------------------------------------------------------------------
__global__ __launch_bounds__(256) void pd_init(
    const float* __restrict__ img, const float* __restrict__ w1p,
    const float* __restrict__ w2p, float* __restrict__ x,
    float* __restrict__ xt, float* __restrict__ y0, float* __restrict__ y1,
    float* __restrict__ wa, float* __restrict__ wb) {
    const unsigned c = blockIdx.x * blockDim.x + threadIdx.x;  // exact grid
    const unsigned i = c >> NC_SHIFT;
    const unsigned jc = c & NC_MASK;
    const unsigned base = (i << 11) + (jc << 2);  // i*NN + jc*4

    const unsigned xoff = (jc < NC - 1) ? 4u : 3u;   // last col -> reload v.w
    const unsigned roff = (i < MM - 1) ? NN : 0u;    // last row -> reload self

    __builtin_prefetch(img + base + roff, 0, 3);     // gfx1250 global_prefetch_b8

    const float4 v = *(const float4*)(img + base);
    const float xn = img[base + xoff];
    const float4 b = *(const float4*)(img + base + roff);

    const float gh0 = v.y - v.x, gh1 = v.z - v.y, gh2 = v.w - v.z, gh3 = xn - v.w;
    const float gv0 = b.x - v.x, gv1 = b.y - v.y, gv2 = b.z - v.z, gv3 = b.w - v.w;

    const float w1s = w1p[0], w2s = w2p[0];
    float4 o;
    o.x = gh0; o.y = gh1; o.z = gh2; o.w = gh3;
    *(float4*)(y0 + base) = o;
    o.x = gv0; o.y = gv1; o.z = gv2; o.w = gv3;
    *(float4*)(y1 + base) = o;

    o.x = w1s + w2s * expf(-fabsf(gh0));
    o.y = w1s + w2s * expf(-fabsf(gh1));
    o.z = w1s + w2s * expf(-fabsf(gh2));
    o.w = w1s + w2s * expf(-fabsf(gh3));
    *(float4*)(wa + base) = o;
    o.x = w1s + w2s * expf(-fabsf(gv0));
    o.y = w1s + w2s * expf(-fabsf(gv1));
    o.z = w1s + w2s * expf(-fabsf(gv2));
    o.w = w1s + w2s * expf(-fabsf(gv3));
    *(float4*)(wb + base) = o;

    *(float4*)(x + base) = v;
    *(float4*)(xt + base) = v;
}

// ---------------------------------------------------------------------------
// dual: y = clamp(y + SIGMA * w * forward_grad(x_tilde), -1, 1)   (branchless)
// ---------------------------------------------------------------------------
__global__ __launch_bounds__(256) void pd_dual(
    const float* __restrict__ xt, const float* __restrict__ wa,
    const float* __restrict__ wb, float* __restrict__ y0,
    float* __restrict__ y1) {
    const unsigned c = blockIdx.x * blockDim.x + threadIdx.x;  // exact grid
    const unsigned i = c >> NC_SHIFT;
    const unsigned jc = c & NC_MASK;
    const unsigned base = (i << 11) + (jc << 2);

    const unsigned xoff = (jc < NC - 1) ? 4u : 3u;
    const unsigned roff = (i < MM - 1) ? NN : 0u;

    __builtin_prefetch(xt + base + roff, 0, 3);      // halo row

    const float4 v = *(const float4*)(xt + base);
    const float xn = xt[base + xoff];
    const float4 b = *(const float4*)(xt + base + roff);

    const float gh0 = v.y - v.x, gh1 = v.z - v.y, gh2 = v.w - v.z, gh3 = xn - v.w;
    const float gv0 = b.x - v.x, gv1 = b.y - v.y, gv2 = b.z - v.z, gv3 = b.w - v.w;

    const float4 wav = *(const float4*)(wa + base);
    const float4 wbv = *(const float4*)(wb + base);
    float4 ya = *(float4*)(y0 + base);
    float4 yb = *(float4*)(y1 + base);

    ya.x = clamp1(ya.x + kSIGMA * wav.x * gh0);
    ya.y = clamp1(ya.y + kSIGMA * wav.y * gh1);
    ya.z = clamp1(ya.z + kSIGMA * wav.z * gh2);
    ya.w = clamp1(ya.w + kSIGMA * wav.w * gh3);
    yb.x = clamp1(yb.x + kSIGMA * wbv.x * gv0);
    yb.y = clamp1(yb.y + kSIGMA * wbv.y * gv1);
    yb.z = clamp1(yb.z + kSIGMA * wbv.z * gv2);
    yb.w = clamp1(yb.w + kSIGMA * wbv.w * gv3);

    *(float4*)(y0 + base) = ya;
    *(float4*)(y1 + base) = yb;
}

// ---------------------------------------------------------------------------
// primal: x_new = (x + TAU*div(w*y) + LT*img) / 1.04 ; xt = x_new + THETA*(x_new-x)
// Branchless: y0 last column and y1 last row are exactly 0 for all iterations
// (init writes 0; clamp(0 + sigma*w*0) == 0), so the reference's right/bottom
// divergence special cases reduce to the interior formula. Left/top edges use
// clamped neighbor addresses + a 0/1 mask.
// ---------------------------------------------------------------------------
__global__ __launch_bounds__(256) void pd_primal(
    const float* __restrict__ img, const float* __restrict__ wa,
    const float* __restrict__ wb, const float* __restrict__ y0,
    const float* __restrict__ y1, float* __restrict__ x,
    float* __restrict__ xt) {
    const unsigned c = blockIdx.x * blockDim.x + threadIdx.x;  // exact grid
    const unsigned i = c >> NC_SHIFT;
    const unsigned jc = c & NC_MASK;
    const unsigned base = (i << 11) + (jc << 2);

    const unsigned loff = (jc > 0) ? 1u : 0u;     // left neighbor (clamped)
    const unsigned uoff = (i > 0) ? NN : 0u;      // up row (clamped)
    const float bL = (jc > 0) ? 1.0f : 0.0f;      // zero-mask at col 0
    const float bU = (i > 0) ? 1.0f : 0.0f;       // zero-mask at row 0

    __builtin_prefetch(y1 + base - uoff, 0, 3);   // halo row

    // q0 = w0*y0 (horizontal)
    const float4 wav = *(const float4*)(wa + base);
    const float4 y0v = *(const float4*)(y0 + base);
    const float q0x = wav.x * y0v.x, q0y = wav.y * y0v.y;
    const float q0z = wav.z * y0v.z, q0w = wav.w * y0v.w;
    const float q0l = wa[base - loff] * y0[base - loff];

    // d_h: interior everywhere; col0 masked; col N-1 exact via q0w==0
    const float dh0 = q0x - bL * q0l;
    const float dh1 = q0y - q0x;
    const float dh2 = q0z - q0y;
    const float dh3 = q0w - q0z;

    // q1 = w1*y1 (vertical)
    const float4 wbv = *(const float4*)(wb + base);
    const float4 y1v = *(const float4*)(y1 + base);
    const float q1x = wbv.x * y1v.x, q1y = wbv.y * y1v.y;
    const float q1z = wbv.z * y1v.z, q1w = wbv.w * y1v.w;
    const float4 wbu = *(const float4*)(wb + base - uoff);
    const float4 y1u = *(const float4*)(y1 + base - uoff);

    // d_v: interior everywhere; row0 masked; row M-1 exact via q1==0
    const float dv0 = q1x - bU * (wbu.x * y1u.x);
    const float dv1 = q1y - bU * (wbu.y * y1u.y);
    const float dv2 = q1z - bU * (wbu.z * y1u.z);
    const float dv3 = q1w - bU * (wbu.w * y1u.w);

    const float4 xv = *(const float4*)(x + base);
    const float4 gv = *(const float4*)(img + base);

    float4 xnew, xtv;
    xnew.x = (xv.x + kTAU * (dh0 + dv0) + kLT * gv.x) / kDEN;
    xnew.y = (xv.y + kTAU * (dh1 + dv1) + kLT * gv.y) / kDEN;
    xnew.z = (xv.z + kTAU * (dh2 + dv2) + kLT * gv.z) / kDEN;
    xnew.w = (xv.w + kTAU * (dh3 + dv3) + kLT * gv.w) / kDEN;
    xtv.x = xnew.x + kTHETA * (xnew.x - xv.x);
    xtv.y = xnew.y + kTHETA * (xnew.y - xv.y);
    xtv.z = xnew.z + kTHETA * (xnew.z - xv.z);
    xtv.w = xnew.w + kTHETA * (xnew.w - xv.w);

    *(float4*)(x + base) = xnew;
    *(float4*)(xt + base) = xtv;
}

// ---------------------------------------------------------------------------
extern "C" void kernel_launch(void* const* d_in, const int* in_sizes, int n_in,
                              void* d_out, int out_size, void* d_ws,
                              size_t ws_size, hipStream_t stream) {
    (void)in_sizes; (void)n_in; (void)out_size; (void)ws_size;
    const float* img = (const float*)d_in[0];
    const float* w1p = (const float*)d_in[1];
    const float* w2p = (const float*)d_in[2];

    float* xt = (float*)d_out;  // x_tilde lives in the output buffer (it IS the output)

    char* ws = (char*)d_ws;     // 5 fp32 planes = 80 MiB of scratch
    float* x  = (float*)(ws + 0 * PLANE_BYTES);
    float* y0 = (float*)(ws + 1 * PLANE_BYTES);
    float* y1 = (float*)(ws + 2 * PLANE_BYTES);
    float* wa = (float*)(ws + 3 * PLANE_BYTES);
    float* wb = (float*)(ws + 4 * PLANE_BYTES);

    const int chunks = MM * NC;            // 1,048,576 float4's (exact multiple of 256)
    const dim3 blk(256);                   // 8 wave32 per block
    const dim3 grd(chunks / 256);          // 4096 blocks, no remainder guard

    pd_init<<<grd, blk, 0, stream>>>(img, w1p, w2p, x, xt, y0, y1, wa, wb);
    for (int it = 0; it < 10; ++it) {
        pd_dual<<<grd, blk, 0, stream>>>(xt, wa, wb, y0, y1);
        pd_primal<<<grd, blk, 0, stream>>>(img, wa, wb, y0, y1, x, xt);
    }
}